// SpaFormer_386547057196
// MI455X (gfx1250) — compile-verified
//
#include <hip/hip_runtime.h>
#include <hip/hip_bf16.h>
#include <math.h>
#include <stdint.h>

// ---------------- problem constants ----------------
constexpr int Bq = 4, Nn = 512, Tt = 12, Cc = 64, Hh = 8, DHd = 8, FFd = 256;
constexpr int Rrows = Bq * Nn * Tt;      // 24576 flattened (b,n,t) rows
constexpr int QKVLD = 3 * Cc;            // 192: fused q|k|v row

// ---------------- types ----------------
typedef float v2f __attribute__((ext_vector_type(2)));
typedef float v8f __attribute__((ext_vector_type(8)));
typedef unsigned int u32x4 __attribute__((ext_vector_type(4)));
typedef unsigned int u32x8 __attribute__((ext_vector_type(8)));
typedef __attribute__((address_space(3))) void LDSvoid;

// ---------------- WMMA f32 16x16x4 ----------------
__device__ __forceinline__ v8f wmma_f32_k4(v2f a, v2f b, v8f c) {
  // D = A(16x4) * B(4x16) + C ; fp32 end to end (CDNA5-only f32 WMMA)
  return __builtin_amdgcn_wmma_f32_16x16x4_f32(false, a, false, b, (short)0, c,
                                               false, false);
}

// ---------------- LDS byte offset of a __shared__ object ----------------
__device__ __forceinline__ uint32_t lds_byte_off(void* p) {
  return (uint32_t)(uintptr_t)(LDSvoid*)p;   // addrspacecast generic->LDS, then ptrtoint
}

// ---------------- Tensor Data Mover: 2D tile load Global->LDS ----------------
// tile: tile_d0 contiguous elements (fp32) per row, tile_d1 rows, row stride
// stride_d0 (fp32 units).  g1_flags carries data_size/pad bits for group1 dword0.
__device__ __forceinline__ void tdm_load_2d(uint64_t gaddr, uint32_t laddr,
                                            uint32_t tile_d0, uint32_t tile_d1,
                                            uint32_t stride_d0,
                                            uint32_t g1_flags) {
  u32x4 g0;
  g0[0] = 1u;                                           // count=1, user desc
  g0[1] = laddr;                                        // lds_addr (bytes)
  g0[2] = (uint32_t)(gaddr & 0xFFFFFFFFu);              // global_addr[31:0]
  g0[3] = (uint32_t)((gaddr >> 32) & 0x1FFFFFFu)        // global_addr[56:32]
          | (2u << 30);                                 // type = 2 ("image")
  u32x8 g1;
  g1[0] = g1_flags;                       // workgroup_mask=0 | data_size | pad cfg
  g1[1] = (tile_d0 & 0xFFFFu) << 16;      // tensor_dim0[15:0]  in bits[63:48]... -> [31:16]
  g1[2] = (tile_d1 & 0xFFFFu) << 16;      // tensor_dim0 hi=0 ; tensor_dim1 lo16
  g1[3] = (tile_d0 & 0xFFFFu) << 16;      // tensor_dim1 hi=0 ; tile_dim0
  g1[4] = (tile_d1 & 0xFFFFu);            // tile_dim1 ; tile_dim2=0 (2D)
  g1[5] = stride_d0;                      // tensor_dim0_stride[31:0]
  g1[6] = 0u;                             // stride hi16=0 ; tensor_dim1_stride lo16=0
  g1[7] = 0u;
  asm volatile("tensor_load_to_lds %0, %1" : : "s"(g0), "s"(g1) : "memory");
}

// ---------------- shuffle reductions (wave32) ----------------
__device__ __forceinline__ float redmax16(float v) {
  for (int m = 1; m < 16; m <<= 1) v = fmaxf(v, __shfl_xor(v, m, 32));
  return v;
}
__device__ __forceinline__ float redsum16(float v) {
  for (int m = 1; m < 16; m <<= 1) v += __shfl_xor(v, m, 32);
  return v;
}
__device__ __forceinline__ float redsum32(float v) {
  for (int m = 1; m < 32; m <<= 1) v += __shfl_xor(v, m, 32);
  return v;
}

// ---------------- x = query + id_emb + in_deg[g_id] + out_deg[g_od] ----------------
__global__ __launch_bounds__(256) void prep_x_kernel(
    const float* __restrict__ q, const float* __restrict__ id_emb,
    const float* __restrict__ ind, const float* __restrict__ outd,
    const int* __restrict__ gid, const int* __restrict__ god,
    float* __restrict__ x, int total) {
  int i = blockIdx.x * 256 + threadIdx.x;
  if (i >= total) return;
  int c = i & (Cc - 1);
  int r = i >> 6;                // row = ((b*N + n)*T + t)
  int n = (r / Tt) % Nn;
  x[i] = q[i] + id_emb[n * Cc + c] + ind[gid[n] * Cc + c] + outd[god[n] * Cc + c];
}

// ---------------- biasM[h][n][m] = rel_emb[adj[n][m]][h] ----------------
__global__ __launch_bounds__(256) void prep_bias_kernel(
    const int* __restrict__ adj, const float* __restrict__ rel,
    float* __restrict__ biasM, int total) {
  int i = blockIdx.x * 256 + threadIdx.x;
  if (i >= total) return;
  int h = i / (Nn * Nn);
  int nm = i % (Nn * Nn);
  biasM[i] = rel[adj[nm] * Hh + h];
}

// ---------------- fused QKV projection: one A-load feeds 3 WMMA chains ----------------
// q|k|v written into fused [R,192] buffer; K = Nc = 64.
__global__ __launch_bounds__(128) void gemm_qkv_kernel(
    const float* __restrict__ A, const float* __restrict__ Wq,
    const float* __restrict__ Wk, const float* __restrict__ Wv,
    float* __restrict__ qkv, int M) {
  int wv = threadIdx.x >> 5, lane = threadIdx.x & 31;
  int lhi = lane >> 4, l16 = lane & 15;
  int tiles = (M >> 4) * (Cc >> 4);
  int wt = blockIdx.x * 4 + wv;
  if (wt >= tiles) return;
  int mt = wt >> 2, nt = wt & 3;
  int row = mt * 16 + l16;
  int col = nt * 16 + l16;
  v8f aq = {}, ak = {}, av = {};
  const float* Ar = A + (long)row * Cc + 2 * lhi;
  for (int k0 = 0; k0 < Cc; k0 += 4) {
    v2f a = *(const v2f*)(Ar + k0);
    int ka = k0 + 2 * lhi;
    v2f bq, bk, bv;
    bq.x = Wq[ka * Cc + col]; bq.y = Wq[(ka + 1) * Cc + col];
    bk.x = Wk[ka * Cc + col]; bk.y = Wk[(ka + 1) * Cc + col];
    bv.x = Wv[ka * Cc + col]; bv.y = Wv[(ka + 1) * Cc + col];
    aq = wmma_f32_k4(a, bq, aq);
    ak = wmma_f32_k4(a, bk, ak);
    av = wmma_f32_k4(a, bv, av);
  }
#pragma unroll
  for (int v = 0; v < 8; ++v) {
    int r = mt * 16 + v + 8 * lhi;
    long base = (long)r * QKVLD + nt * 16 + l16;
    qkv[base]       = aq[v];
    qkv[base + 64]  = ak[v];
    qkv[base + 128] = av[v];
  }
}

// ---------------- generic fp32 WMMA GEMM: C = A[M,K] @ B[K,Nc] (+bias)(+relu) ----------------
__global__ __launch_bounds__(128) void gemm_wmma_kernel(
    const float* __restrict__ A, const float* __restrict__ Bm,
    const float* __restrict__ bias, float* __restrict__ C,
    int M, int K, int Nc, int ldc, int relu) {
  int wv = threadIdx.x >> 5, lane = threadIdx.x & 31;
  int lhi = lane >> 4, l16 = lane & 15;
  int ntn = Nc >> 4;
  int tiles = (M >> 4) * ntn;
  int wt = blockIdx.x * 4 + wv;
  if (wt >= tiles) return;           // whole wave exits: EXEC stays all-ones
  int mt = wt / ntn, nt = wt % ntn;
  int row = mt * 16 + l16;           // A: lane holds M=row
  int col = nt * 16 + l16;           // B/C: lane holds N=col
  v8f acc = {};
  const float* Ar = A + (long)row * K + 2 * lhi;
  for (int k0 = 0; k0 < K; k0 += 4) {
    v2f av = *(const v2f*)(Ar + k0);                       // A[row][k0+2*lhi ..+1]
    int ka = k0 + 2 * lhi;
    v2f bv;
    bv.x = Bm[(long)ka * Nc + col];                        // B[ka  ][col]
    bv.y = Bm[(long)(ka + 1) * Nc + col];                  // B[ka+1][col]
    acc = wmma_f32_k4(av, bv, acc);
  }
#pragma unroll
  for (int v = 0; v < 8; ++v) {
    int r = mt * 16 + v + 8 * lhi;
    float val = acc[v] + (bias ? bias[col] : 0.f);
    if (relu) val = fmaxf(val, 0.f);
    C[(long)r * ldc + col] = val;
  }
}

// ---------------- fused flash attention over nodes, per (b,h,t) ----------------
__global__ __launch_bounds__(256) void attn_kernel(
    const float* __restrict__ qkv, const float* __restrict__ biasM,
    float* __restrict__ ctx) {
  __shared__ float Ksh[Nn * 8];        // K [512][8]   (TDM tile, no padding)
  __shared__ float Vsh[Nn * 16];       // V [512][16]  (TDM tile, pad 8 dw per 8 dw)
  __shared__ float Psc[8][16 * 16];    // per-wave P-tile transpose scratch

  int bid = blockIdx.x;                // B*H*T = 384 blocks
  int b = bid / (Hh * Tt);
  int rem = bid % (Hh * Tt);
  int h = rem / Tt, t = rem % Tt;
  int tid = threadIdx.x;

  // zero the padded V columns 8..15 (TDM skips them via pad_amount)
  for (int idx = tid; idx < Nn * 4; idx += 256) {
    int m = idx >> 2;
    int d0 = (idx & 3) * 2;
    v2f z; z.x = 0.f; z.y = 0.f;
    *(v2f*)(Vsh + m * 16 + 8 + d0) = z;
  }

  // wave 0 drives the Tensor Data Mover for both K and V tiles:
  // tile_dim0 = 8 floats (contiguous d), tile_dim1 = 512 rows (m),
  // row stride = T*192 = 2304 floats.
  if (tid < 32) {
    long base0 = ((long)(b * Nn) * Tt + t) * QKVLD + h * DHd;
    uint64_t kga = (uint64_t)(uintptr_t)(qkv + base0 + 64);
    uint64_t vga = (uint64_t)(uintptr_t)(qkv + base0 + 128);
    uint32_t dsz4 = 2u << 16;                       // data_size = 4 bytes
    // V: pad_enable | pad_interval=2 (8 DWORDs) | pad_amount=7 (8 DWORDs)
    uint32_t padv = dsz4 | (1u << 20) | (2u << 22) | (7u << 25);
    tdm_load_2d(kga, lds_byte_off(Ksh), DHd, Nn, (uint32_t)(Tt * QKVLD), dsz4);
    tdm_load_2d(vga, lds_byte_off(Vsh), DHd, Nn, (uint32_t)(Tt * QKVLD), padv);
    __builtin_amdgcn_s_wait_tensorcnt(0);           // both DMAs done
  }
  __syncthreads();

  int wv = tid >> 5, lane = tid & 31;
  int lhi = lane >> 4, l16 = lane & 15;
  const float scale = 0.35355339059327373f;   // 1/sqrt(DH=8)
  const float* bh = biasM + (long)h * Nn * Nn;

  for (int rt = wv; rt < Nn / 16; rt += 8) {  // 32 row tiles / 8 waves
    int rowbase = rt * 16;
    // Q fragments (reused across all column tiles): A-layout, lane=M, K-pair=2*lhi
    int nq = rowbase + l16;
    long qb = ((long)(b * Nn + nq) * Tt + t) * QKVLD + h * DHd;
    v2f qa0 = *(const v2f*)(qkv + qb + 2 * lhi);       // d = 0..3
    v2f qa1 = *(const v2f*)(qkv + qb + 4 + 2 * lhi);   // d = 4..7

    v8f oacc = {};
    float mrun[8], lrun[8];
#pragma unroll
    for (int v = 0; v < 8; ++v) { mrun[v] = -3.0e38f; lrun[v] = 0.f; }

    for (int ct = 0; ct < Nn / 16; ++ct) {
      int mcol = ct * 16 + l16;
      if (ct + 1 < Nn / 16)   // prefetch next bias tile line (global_prefetch_b8)
        __builtin_prefetch(bh + (long)(rowbase + 8 * lhi) * Nn + mcol + 16, 0, 0);
      // scores tile = Q(16x8) @ K^T(8x16), two chained K=4 WMMAs
      v2f kb0 = *(const v2f*)(Ksh + mcol * 8 + 2 * lhi);
      v2f kb1 = *(const v2f*)(Ksh + mcol * 8 + 4 + 2 * lhi);
      v8f s = {};
      s = wmma_f32_k4(qa0, kb0, s);
      s = wmma_f32_k4(qa1, kb1, s);

      float p[8], alpha[8];
#pragma unroll
      for (int v = 0; v < 8; ++v) {
        int nrow = rowbase + v + 8 * lhi;
        float sv = s[v] * scale + bh[(long)nrow * Nn + mcol];
        float mx = redmax16(sv);                    // row max over 16 lanes
        float mnew = fmaxf(mrun[v], mx);
        float pv = __expf(sv - mnew);
        float ls = redsum16(pv);
        float a = __expf(mrun[v] - mnew);
        lrun[v] = lrun[v] * a + ls;
        mrun[v] = mnew;
        alpha[v] = a;
        p[v] = pv;
      }
      // rescale running output, spill P tile (C-layout) to LDS for transpose
#pragma unroll
      for (int v = 0; v < 8; ++v) {
        oacc[v] *= alpha[v];
        Psc[wv][(v + 8 * lhi) * 16 + l16] = p[v];
      }
      // oacc += P(16x16) @ V(16x16pad), four chained K=4 WMMAs
#pragma unroll
      for (int c4 = 0; c4 < 4; ++c4) {
        v2f pa = *(const v2f*)(&Psc[wv][l16 * 16 + c4 * 4 + 2 * lhi]);
        int m0 = ct * 16 + c4 * 4 + 2 * lhi;
        v2f vb;
        vb.x = Vsh[m0 * 16 + l16];
        vb.y = Vsh[(m0 + 1) * 16 + l16];
        oacc = wmma_f32_k4(pa, vb, oacc);
      }
    }
    // write ctx[b,n,t, h*8+d] = o / l  (cols 0..7 valid)
    if (l16 < DHd) {
#pragma unroll
      for (int v = 0; v < 8; ++v) {
        int nrow = rowbase + v + 8 * lhi;
        ctx[((long)(b * Nn + nrow) * Tt + t) * Cc + h * DHd + l16] =
            oacc[v] / lrun[v];
      }
    }
  }
}

// ---------------- LayerNorm(a + res) * g + b, one wave per 64-wide row ----------------
__global__ __launch_bounds__(256) void ln_kernel(
    const float* __restrict__ a, const float* __restrict__ res,
    const float* __restrict__ g, const float* __restrict__ bt,
    float* __restrict__ out, int R) {
  int row = blockIdx.x * 8 + (threadIdx.x >> 5);
  if (row >= R) return;
  int lane = threadIdx.x & 31;
  const float* pa = a + (long)row * Cc;
  const float* pr = res + (long)row * Cc;
  float v0 = pa[lane] + pr[lane];
  float v1 = pa[lane + 32] + pr[lane + 32];
  float mu = redsum32(v0 + v1) * (1.f / 64.f);
  float d0 = v0 - mu, d1 = v1 - mu;
  float var = redsum32(d0 * d0 + d1 * d1) * (1.f / 64.f);
  float inv = rsqrtf(var + 1e-5f);
  out[(long)row * Cc + lane]      = d0 * inv * g[lane] + bt[lane];
  out[(long)row * Cc + lane + 32] = d1 * inv * g[lane + 32] + bt[lane + 32];
}

// ---------------- host orchestration ----------------
extern "C" void kernel_launch(void* const* d_in, const int* in_sizes, int n_in,
                              void* d_out, int out_size, void* d_ws, size_t ws_size,
                              hipStream_t stream) {
  (void)in_sizes; (void)n_in; (void)out_size; (void)ws_size;
  const float* query = (const float*)d_in[0];
  const int*   adj   = (const int*)d_in[1];
  const int*   g_id  = (const int*)d_in[2];
  const int*   g_od  = (const int*)d_in[3];
  const float* id_e  = (const float*)d_in[4];
  const float* ind_e = (const float*)d_in[5];
  const float* oud_e = (const float*)d_in[6];
  const float* rel   = (const float*)d_in[7];
  const float* wq    = (const float*)d_in[8];
  const float* wk    = (const float*)d_in[9];
  const float* wvw   = (const float*)d_in[10];
  const float* wo    = (const float*)d_in[11];
  const float* bo    = (const float*)d_in[12];
  const float* ln1g  = (const float*)d_in[13];
  const float* ln1b  = (const float*)d_in[14];
  const float* ln2g  = (const float*)d_in[15];
  const float* ln2b  = (const float*)d_in[16];
  const float* w1    = (const float*)d_in[17];
  const float* bf1   = (const float*)d_in[18];
  const float* w2    = (const float*)d_in[19];
  const float* bf2   = (const float*)d_in[20];
  float* out = (float*)d_out;

  float* ws = (float*)d_ws;
  size_t o = 0;
  float* x     = ws + o; o += (size_t)Rrows * Cc;        // residual input
  float* biasM = ws + o; o += (size_t)Hh * Nn * Nn;      // [H,N,N]
  float* qkv   = ws + o; o += (size_t)Rrows * QKVLD;     // fused q|k|v
  float* ctx   = ws + o; o += (size_t)Rrows * Cc;
  float* att   = ws + o; o += (size_t)Rrows * Cc;
  float* xs    = ws + o; o += (size_t)Rrows * Cc;
  float* ffh   = ws + o; o += (size_t)Rrows * FFd;
  float* ff2o  = ws + o; o += (size_t)Rrows * Cc;

  // 1. embeddings fused into x
  prep_x_kernel<<<(Rrows * Cc) / 256, 256, 0, stream>>>(
      query, id_e, ind_e, oud_e, g_id, g_od, x, Rrows * Cc);
  // 2. attention bias [H,N,N]
  prep_bias_kernel<<<(Hh * Nn * Nn) / 256, 256, 0, stream>>>(
      adj, rel, biasM, Hh * Nn * Nn);
  // 3. fused QKV projection (one pass over x)
  int tiles64 = (Rrows / 16) * (Cc / 16);
  gemm_qkv_kernel<<<(tiles64 + 3) / 4, 128, 0, stream>>>(
      x, wq, wk, wvw, qkv, Rrows);
  // 4. fused flash attention (one WG per (b,h,t), TDM-staged K/V)
  attn_kernel<<<Bq * Hh * Tt, 256, 0, stream>>>(qkv, biasM, ctx);
  // 5. output projection
  gemm_wmma_kernel<<<(tiles64 + 3) / 4, 128, 0, stream>>>(
      ctx, wo, bo, att, Rrows, Cc, Cc, Cc, 0);
  // 6. LN1(att + x)
  ln_kernel<<<(Rrows + 7) / 8, 256, 0, stream>>>(att, x, ln1g, ln1b, xs, Rrows);
  // 7. FF1 + ReLU
  int tilesFF1 = (Rrows / 16) * (FFd / 16);
  gemm_wmma_kernel<<<(tilesFF1 + 3) / 4, 128, 0, stream>>>(
      xs, w1, bf1, ffh, Rrows, Cc, FFd, FFd, 1);
  // 8. FF2
  gemm_wmma_kernel<<<(tiles64 + 3) / 4, 128, 0, stream>>>(
      ffh, w2, bf2, ff2o, Rrows, FFd, Cc, Cc, 0);
  // 9. LN2(ff2 + xs) -> out
  ln_kernel<<<(Rrows + 7) / 8, 256, 0, stream>>>(ff2o, xs, ln2g, ln2b, out, Rrows);
}